// DeformConvWarpNormSingle_74251394613307
// MI455X (gfx1250) — compile-verified
//
#include <hip/hip_runtime.h>

// Problem constants (from reference setup_inputs)
#define BB   2
#define CC   64
#define HH   256
#define WW   256
#define K2C  9

// Tiling
#define TPX   16              // output pixels per tile (one WMMA M dim)
#define WIN_H 11              // halo rows:  h-5 .. h+5 (center-out slot order)
#define WIN_W 26              // halo cols:  w0-5 .. w0+20
#define NSRC  (WIN_H * WIN_W) // 286 window sources
#define STR   292             // LDS row stride (words): breaks stride-64-bank pattern

#if defined(__gfx1250__) && __has_builtin(__builtin_amdgcn_global_load_async_to_lds_b32)
#define USE_ASYNC_LDS 1
#endif

typedef __attribute__((ext_vector_type(2))) float v2f;
typedef __attribute__((ext_vector_type(8))) float v8f;
typedef __attribute__((address_space(1))) int gas_int;   // global AS pointer pointee
typedef __attribute__((address_space(3))) int las_int;   // LDS AS pointer pointee

// Center-out row permutation: physical window row (0..10, 5=center) <-> slot.
__device__ __forceinline__ int rowSlot(int dy) {           // window row -> slot
    return (dy == 5) ? 0 : (dy < 5 ? (2 * (5 - dy) - 1) : (2 * (dy - 5)));
}
__device__ __forceinline__ int slotRow(int s) {            // slot -> window row
    return (s & 1) ? (5 - ((s + 1) >> 1)) : (5 + (s >> 1));
}

__global__ __launch_bounds__(256)
void dcn_wmma_kernel(const float* __restrict__ xg,
                     const float* __restrict__ offs,
                     const float* __restrict__ maskp,
                     const float* __restrict__ wq,
                     float* __restrict__ outp)
{
    extern __shared__ float smem[];
    float* Xw = smem;                 // [CC][STR]   staged window, B matrix source
    float* As = Xw + CC * STR;        // [TPX][STR]  sparse coefficient matrix A
    float* Fs = As + TPX * STR;       // [TPX][CC]   exact fallback accumulator
    __shared__ int slotMaxSh;         // max center-out slot touched this tile

    const int tid  = threadIdx.x;
    const int tile = blockIdx.x;
    const int tilesPerRow = WW / TPX;                 // 16
    const int b   = tile / (HH * tilesPerRow);
    const int rem = tile % (HH * tilesPerRow);
    const int h   = rem / tilesPerRow;
    const int w0  = (rem % tilesPerRow) * TPX;
    const int y_lo = h - 5;
    const int x_lo = w0 - 5;

    // ---- zero A and F (contiguous in LDS), init slot bound ----
    for (int i = tid; i < TPX * STR + TPX * CC; i += 256)
        As[i] = 0.0f;
    if (tid == 0) slotMaxSh = 0;
    __syncthreads();

    // ---- phase 1: build A in center-out slot space (rare exact fallback -> F) ----
    if (tid < TPX * K2C) {
        const int p = tid / K2C;
        const int k = tid % K2C;
        const int w = w0 + p;
        const int ki = k / 3;
        const int kj = k % 3;

        const float offy = offs[(((b * 2 * K2C + 2 * k    ) * HH + h) << 8) + w];
        const float offx = offs[(((b * 2 * K2C + 2 * k + 1) * HH + h) << 8) + w];
        const float mk   = maskp[(((b * K2C + k) * HH + h) << 8) + w] * wq[k];

        const float py = (float)(h - 1 + ki) + offy;
        const float px = (float)(w - 1 + kj) + offx;
        const float y0f = floorf(py);
        const float x0f = floorf(px);
        const float ly = py - y0f;
        const float lx = px - x0f;
        const int iy0 = (int)y0f;
        const int ix0 = (int)x0f;

        const float w00 = (1.0f - ly) * (1.0f - lx) * mk;
        const float w01 = (1.0f - ly) * lx          * mk;
        const float w10 = ly          * (1.0f - lx) * mk;
        const float w11 = ly          * lx          * mk;

        const bool inWin = (iy0 >= y_lo) && (iy0 + 1 <= h + 5) &&
                           (ix0 >= x_lo) && (ix0 + 1 <= w0 + 20);
        if (inWin) {
            const int ry = iy0 - y_lo;           // 0..9
            const int rx = ix0 - x_lo;           // 0..24
            const int s0 = rowSlot(ry);
            const int s1 = rowSlot(ry + 1);
            const int j0 = s0 * WIN_W + rx;
            const int j1 = s1 * WIN_W + rx;
            atomicAdd(&As[p * STR + j0    ], w00);
            atomicAdd(&As[p * STR + j0 + 1], w01);
            atomicAdd(&As[p * STR + j1    ], w10);
            atomicAdd(&As[p * STR + j1 + 1], w11);
            atomicMax(&slotMaxSh, s0 > s1 ? s0 : s1);
        } else {
            // exact fallback for |offset| tail (~1e-4 of taps): direct gather
            const int   cy[4] = { iy0, iy0, iy0 + 1, iy0 + 1 };
            const int   cx[4] = { ix0, ix0 + 1, ix0, ix0 + 1 };
            const float cw[4] = { w00, w01, w10, w11 };
            for (int q = 0; q < 4; q++) {
                const int yc = cy[q], xc = cx[q];
                if (yc >= 0 && yc < HH && xc >= 0 && xc < WW) {
                    const float* xp = xg + ((size_t)b * CC * HH * WW)
                                         + ((size_t)yc << 8) + xc;
                    const float cf = cw[q];
                    for (int c = 0; c < CC; c++)
                        atomicAdd(&Fs[p * CC + c], cf * xp[(size_t)c * HH * WW]);
                }
            }
        }
    }
    __syncthreads();

    // ---- phase 2: stage only the touched slots (coalesced; zeros elsewhere) ----
    const int nAct = (slotMaxSh + 1) * WIN_W;     // active source cells (<= 286)
    const int kHi  = (nAct + 3) & ~3;             // WMMA K bound (mult of 4, <= 288)
    {
        const int c = tid >> 2;                   // 64 channels x 4 lanes each
        const float* cbase = xg + (((size_t)(b * CC + c) * HH) << 8);
        float* lrow = Xw + c * STR;
        for (int j = (tid & 3); j < kHi; j += 4) {
            bool ld = false;
            const float* gp = cbase;
            if (j < nAct) {
                const int s  = j / WIN_W;
                const int cl = j - s * WIN_W;
                const int gy = y_lo + slotRow(s);
                const int gx = x_lo + cl;
                if (gy >= 0 && gy < HH && gx >= 0 && gx < WW) {
                    gp = cbase + (gy << 8) + gx;
                    ld = true;
                }
            }
#ifdef USE_ASYNC_LDS
            if (ld) {
                __builtin_amdgcn_global_load_async_to_lds_b32(
                    (gas_int*)gp, (las_int*)(lrow + j), 0, 0);
            } else {
                lrow[j] = 0.0f;
            }
#else
            lrow[j] = ld ? *gp : 0.0f;
#endif
        }
    }
#ifdef USE_ASYNC_LDS
#if __has_builtin(__builtin_amdgcn_s_wait_asynccnt)
    __builtin_amdgcn_s_wait_asynccnt(0);
#else
    asm volatile("s_wait_asynccnt 0" ::: "memory");
#endif
#endif
    __syncthreads();

    // ---- phase 3: D[16 px, 16 ch] = A(16xK) x Xw(Kx16), K = kHi, f32 WMMA ----
    const int wid = tid >> 5;
    if (wid < 4) {
        const int lane = tid & 31;
        const int ln   = lane & 15;   // A: row M ; B/D: column N
        const int hi   = lane >> 4;   // 0: K pair {0,1}; 1: K pair {2,3}
        const int c0   = wid * 16;    // channel group

        v8f acc = {0.f, 0.f, 0.f, 0.f, 0.f, 0.f, 0.f, 0.f};
        const float* arow = As + ln * STR + 2 * hi;
        const float* brow = Xw + (c0 + ln) * STR + 2 * hi;

        for (int jb = 0; jb < kHi; jb += 4) {
            v2f a  = *(const v2f*)(arow + jb);   // A[M=ln][jb + 2*hi .. +1]
            v2f bf = *(const v2f*)(brow + jb);   // B[K=jb+2*hi ..][N=ln]
            acc = __builtin_amdgcn_wmma_f32_16x16x4_f32(
                    /*neg_a=*/false, a, /*neg_b=*/false, bf,
                    /*c_mod=*/(short)0, acc, /*reuse_a=*/false, /*reuse_b=*/false);
        }

        // D layout: VGPR r holds M=r (lanes 0-15) and M=r+8 (lanes 16-31), N=ln
        #pragma unroll
        for (int r = 0; r < 8; r++) {
            const int p = r + 8 * hi;
            const int c = c0 + ln;
            const float v = acc[r] + Fs[p * CC + c];
            outp[(((b * CC + c) * HH + h) << 8) + (w0 + p)] = v;
        }
    }
}

extern "C" void kernel_launch(void* const* d_in, const int* in_sizes, int n_in,
                              void* d_out, int out_size, void* d_ws, size_t ws_size,
                              hipStream_t stream) {
    (void)in_sizes; (void)n_in; (void)out_size; (void)d_ws; (void)ws_size;
    const float* x    = (const float*)d_in[0];
    const float* off  = (const float*)d_in[1];
    const float* mask = (const float*)d_in[2];
    const float* wt   = (const float*)d_in[3];
    float* out = (float*)d_out;

    const int grid = BB * HH * (WW / TPX);                         // 8192 tiles
    const size_t sh = (size_t)(CC * STR + TPX * STR + TPX * CC) * sizeof(float); // ~97.5 KB
    dcn_wmma_kernel<<<grid, 256, sh, stream>>>(x, off, mask, wt, out);
}